// CannyFilter_5746666242328
// MI455X (gfx1250) — compile-verified
//
#include <hip/hip_runtime.h>
#include <hip/hip_bf16.h>
#include <stdint.h>

#define H_IMG 640
#define W_IMG 640
#define NIMG  24              // 8 batches * 3 channels
#define HW    (H_IMG * W_IMG)

typedef __attribute__((ext_vector_type(2))) float v2f;
typedef __attribute__((ext_vector_type(8))) float v8f;

// normalized 1-D gaussian taps (sigma=1, ks=5); outer(g,g) == reference 2-D kernel
// branchless band select: v_cmp + v_cndmask, no exec manipulation
__device__ __forceinline__ float gcoef(int d) {
    float r = (d == 2) ? 0.40261994689744629f : 0.0f;
    r = (d == 1 || d == 3) ? 0.24420134200312975f : r;
    r = (d == 0 || d == 4) ? 0.05448868454964294f : r;
    return r;
}

// zero-padded image read: clamp address (always legal) + select-zero.
// Lowered as unconditional global_load_b32 + v_cndmask (no saveexec branches).
__device__ __forceinline__ float ldpix(const float* __restrict__ img, int r, int c) {
    bool ok = ((unsigned)r < (unsigned)H_IMG) && ((unsigned)c < (unsigned)W_IMG);
    int rc = r < 0 ? 0 : (r > H_IMG - 1 ? H_IMG - 1 : r);
    int cc = c < 0 ? 0 : (c > W_IMG - 1 ? W_IMG - 1 : c);
    float v = img[(size_t)rc * W_IMG + cc];
    return ok ? v : 0.f;
}

__device__ __forceinline__ unsigned char ldbin(const unsigned char* __restrict__ img,
                                               int r, int c) {
    bool ok = ((unsigned)r < (unsigned)H_IMG) && ((unsigned)c < (unsigned)W_IMG);
    int rc = r < 0 ? 0 : (r > H_IMG - 1 ? H_IMG - 1 : r);
    int cc = c < 0 ? 0 : (c > W_IMG - 1 ? W_IMG - 1 : c);
    unsigned char v = img[(size_t)rc * W_IMG + cc];
    return ok ? v : (unsigned char)0;
}

// -------- K1: separable 5x5 Gaussian via banded-matrix WMMA (f32 16x16x4) -----
// One wave (32 threads) per 16x16 output tile.
//   stage1: T[k][n] = sum_dx g[dx] * x[r0-2+k][c0+n+dx-2]   (horizontal, rows r0-2..r0+17)
//           rows 0..15 via 5 chunked WMMAs:  T = A * B,  A[m,k]=x[..], B[k,n]=g[k-n]
//           rows 16..19 via plain VALU (halo)
//   stage2: out[m][n] = sum_k g[k-m] * T[k][n]              (vertical)
//           5 chunked WMMAs: A[m,k]=g[k-m] (const band), B[k,n]=T from LDS
__global__ __launch_bounds__(32) void k_blur_wmma(const float* __restrict__ x,
                                                  float* __restrict__ blur) {
    const int TILES = W_IMG / 16;           // 40
    int wid = blockIdx.x;
    int img = wid / (TILES * TILES);
    int t   = wid % (TILES * TILES);
    int r0  = (t / TILES) * 16;
    int c0  = (t % TILES) * 16;
    const float* src = x    + (size_t)img * HW;
    float*       dst = blur + (size_t)img * HW;

    int lane = threadIdx.x & 31;
    int half = lane >> 4;                   // 0: lanes 0-15, 1: lanes 16-31
    int lid  = lane & 15;

    __shared__ float ldsT[20][16];

    // ---------------- stage 1: horizontal (WMMA rows r0-2 .. r0+13) ----------
    v8f acc = {0.f, 0.f, 0.f, 0.f, 0.f, 0.f, 0.f, 0.f};
    int ra = r0 - 2 + lid;                  // A-operand row (absolute)
#pragma unroll
    for (int c = 0; c < 5; ++c) {
        int k0  = 4 * c + 2 * half;         // this lane's K slots: k0, k0+1
        int ca0 = c0 - 2 + k0;
        v2f a, b;
        a.x = ldpix(src, ra, ca0);
        a.y = ldpix(src, ra, ca0 + 1);
        b.x = gcoef(k0 - lid);              // banded horizontal weights
        b.y = gcoef(k0 + 1 - lid);
        acc = __builtin_amdgcn_wmma_f32_16x16x4_f32(false, a, false, b,
                                                    (short)0, acc, false, false);
    }
#pragma unroll
    for (int v = 0; v < 8; ++v)             // C layout -> LDS (row = v + 8*half)
        ldsT[v + 8 * half][lid] = acc[v];

    // ---------------- stage 1b: halo rows r0+14 .. r0+17 (local 16..19) ------
#pragma unroll
    for (int i = 0; i < 2; ++i) {
        int id = lane + 32 * i;             // 0..63
        int e  = id >> 4;                   // 0..3
        int n  = id & 15;
        int rr = r0 + 14 + e;
        float s = 0.f;
#pragma unroll
        for (int d = 0; d < 5; ++d)
            s += gcoef(d) * ldpix(src, rr, c0 + n - 2 + d);
        ldsT[16 + e][n] = s;
    }
    __syncthreads();

    // ---------------- stage 2: vertical (WMMA, const banded A) ---------------
    v8f out = {0.f, 0.f, 0.f, 0.f, 0.f, 0.f, 0.f, 0.f};
#pragma unroll
    for (int c = 0; c < 5; ++c) {
        int k0 = 4 * c + 2 * half;
        v2f a, b;
        a.x = gcoef(k0 - lid);              // Wv[m=lid, k0]
        a.y = gcoef(k0 + 1 - lid);
        b.x = ldsT[k0][lid];                // T[k0][n=lid]
        b.y = ldsT[k0 + 1][lid];
        out = __builtin_amdgcn_wmma_f32_16x16x4_f32(false, a, false, b,
                                                    (short)0, out, false, false);
    }
#pragma unroll
    for (int v = 0; v < 8; ++v)
        dst[(size_t)(r0 + v + 8 * half) * W_IMG + c0 + lid] = out[v];
}

// -------- K0: seed per-channel min/max (uint bit-pattern ordering, mag >= 0) --
__global__ void k_init(unsigned int* mm) {
    int t = threadIdx.x;
    if (t < 6) mm[t] = (t < 3) ? 0x7f7fffffu : 0u;  // min=FLT_MAX bits, max=0
}

// -------- K2: Sobel -> mag + angle-bin, per-channel min/max atomics ----------
__global__ __launch_bounds__(256) void k_sobel(const float* __restrict__ blur,
                                               float* __restrict__ mag,
                                               unsigned char* __restrict__ bin,
                                               unsigned int* __restrict__ mm) {
    int img = blockIdx.z;
    int r0 = blockIdx.y * 16, c0 = blockIdx.x * 16;
    const float* b = blur + (size_t)img * HW;
    int tx = threadIdx.x, ty = threadIdx.y;
    int t = ty * 16 + tx;

    __shared__ float bs[18][18];            // blur tile + 1-halo
    for (int i = t; i < 324; i += 256) {
        int a = i / 18, b2 = i % 18;
        bs[a][b2] = ldpix(b, r0 - 1 + a, c0 - 1 + b2);
    }
    __syncthreads();

    float nw = bs[ty][tx],     nn = bs[ty][tx + 1],     ne = bs[ty][tx + 2];
    float ww = bs[ty + 1][tx],                          ee = bs[ty + 1][tx + 2];
    float sw = bs[ty + 2][tx], ss = bs[ty + 2][tx + 1], se = bs[ty + 2][tx + 2];

    float gx = (ne + 2.f * ee + se) - (nw + 2.f * ww + sw);
    float gy = (sw + 2.f * ss + se) - (nw + 2.f * nn + ne);
    float m  = sqrtf(gx * gx + gy * gy);

    // angle bin via tangent thresholds (equivalent to degrees-mod-180 binning)
    float ax = fabsf(gx), ay = fabsf(gy);
    const float T1 = 0.41421356237309503f;   // tan 22.5
    const float T2 = 2.41421356237309510f;   // tan 67.5
    unsigned char bb = ((gx > 0.f) == (gy > 0.f)) ? (unsigned char)1 : (unsigned char)3;
    bb = (ay >= T2 * ax) ? (unsigned char)2 : bb;
    bb = (ay < T1 * ax)  ? (unsigned char)0 : bb;

    size_t idx = (size_t)img * HW + (size_t)(r0 + ty) * W_IMG + (c0 + tx);
    mag[idx] = m;
    bin[idx] = bb;

    __shared__ float smin[256], smax[256];
    smin[t] = m; smax[t] = m;
    __syncthreads();
    for (int s = 128; s > 0; s >>= 1) {
        if (t < s) {
            smin[t] = fminf(smin[t], smin[t + s]);
            smax[t] = fmaxf(smax[t], smax[t + s]);
        }
        __syncthreads();
    }
    if (t == 0) {
        int ch = img % 3;
        atomicMin(&mm[ch],     __float_as_uint(smin[0]));
        atomicMax(&mm[3 + ch], __float_as_uint(smax[0]));
    }
}

// -------- K3: NMS + double threshold + single-pass hysteresis ----------------
// Normalization is per-channel monotone affine -> compare raw mags, map
// thresholds instead: Tlo/Thi = mn + {0.1,0.3}*(mx-mn+1e-8).
__global__ __launch_bounds__(256) void k_nms_hyst(const float* __restrict__ mag,
                                                  const unsigned char* __restrict__ bin,
                                                  const unsigned int* __restrict__ mm,
                                                  float* __restrict__ out) {
    int img = blockIdx.z;
    int r0 = blockIdx.y * 16, c0 = blockIdx.x * 16;
    const float* mg         = mag + (size_t)img * HW;
    const unsigned char* bi = bin + (size_t)img * HW;
    int t = threadIdx.y * 16 + threadIdx.x;

    int ch = img % 3;
    float mn = __uint_as_float(mm[ch]);
    float mx = __uint_as_float(mm[3 + ch]);
    float denom = mx - mn + 1e-8f;
    float Thi = mn + 0.3f * denom;
    float Tlo = mn + 0.1f * denom;

    __shared__ float         ms[20][20];   // mag, 2-halo
    __shared__ unsigned char bs[18][18];   // bins, 1-halo
    __shared__ unsigned char cs[18][18];   // class: 0 none, 1 weak, 2 strong

    for (int i = t; i < 400; i += 256) {
        int a = i / 20, b2 = i % 20;
        ms[a][b2] = ldpix(mg, r0 - 2 + a, c0 - 2 + b2);
    }
    for (int i = t; i < 324; i += 256) {
        int a = i / 18, b2 = i % 18;
        bs[a][b2] = ldbin(bi, r0 - 1 + a, c0 - 1 + b2);
    }
    __syncthreads();

    for (int i = t; i < 324; i += 256) {
        int a = i / 18, b2 = i % 18;             // extended tile coords
        int rr = r0 - 1 + a, cc = c0 - 1 + b2;
        unsigned char cls = 0;
        if (rr >= 1 && rr < H_IMG - 1 && cc >= 1 && cc < W_IMG - 1) {
            float m = ms[a + 1][b2 + 1];
            int bb = bs[a][b2];
            float n1, n2;
            if (bb == 0)      { n1 = ms[a + 1][b2 + 2]; n2 = ms[a + 1][b2];     } // E, W
            else if (bb == 1) { n1 = ms[a + 2][b2];     n2 = ms[a][b2 + 2];     } // SW, NE
            else if (bb == 2) { n1 = ms[a + 2][b2 + 1]; n2 = ms[a][b2 + 1];     } // S, N
            else              { n1 = ms[a + 2][b2 + 2]; n2 = ms[a][b2];         } // SE, NW
            if (m >= n1 && m >= n2) {
                if (m >= Thi)      cls = 2;
                else if (m >= Tlo) cls = 1;
            }
        }
        cs[a][b2] = cls;
    }
    __syncthreads();

    int a = threadIdx.y + 1, b2 = threadIdx.x + 1;
    unsigned char c = cs[a][b2];
    float e = 0.f;
    if (c == 2) e = 1.f;
    else if (c == 1) {
        bool any = false;
#pragma unroll
        for (int dr = -1; dr <= 1; ++dr)
#pragma unroll
            for (int dc = -1; dc <= 1; ++dc)
                any = any || (cs[a + dr][b2 + dc] == 2);
        if (any) e = 1.f;
    }
    out[(size_t)img * HW + (size_t)(r0 + threadIdx.y) * W_IMG + (c0 + threadIdx.x)] = e;
}

extern "C" void kernel_launch(void* const* d_in, const int* in_sizes, int n_in,
                              void* d_out, int out_size, void* d_ws, size_t ws_size,
                              hipStream_t stream) {
    const float* x = (const float*)d_in[0];
    float* out = (float*)d_out;

    char* w = (char*)d_ws;
    const size_t N = (size_t)NIMG * HW;                 // 9,830,400 elems
    float*         blur = (float*)w;                    // N * 4 bytes
    float*         mag  = (float*)(w + N * 4);          // N * 4 bytes
    unsigned char* bin  = (unsigned char*)(w + N * 8);  // N bytes
    unsigned int*  mm   = (unsigned int*)(w + N * 8 + ((N + 255) & ~(size_t)255));

    hipLaunchKernelGGL(k_init, dim3(1), dim3(32), 0, stream, mm);
    hipLaunchKernelGGL(k_blur_wmma, dim3(NIMG * 40 * 40), dim3(32), 0, stream, x, blur);
    dim3 g(40, 40, NIMG), b(16, 16);
    hipLaunchKernelGGL(k_sobel, g, b, 0, stream, blur, mag, bin, mm);
    hipLaunchKernelGGL(k_nms_hyst, g, b, 0, stream, mag, bin, mm, out);
}